// MultiHeadAttention_27238682591285
// MI455X (gfx1250) — compile-verified
//
#include <hip/hip_runtime.h>

typedef __attribute__((ext_vector_type(2))) float v2f;
typedef __attribute__((ext_vector_type(8))) float v8f;

#define WMMA_F32(a, b, c) \
  __builtin_amdgcn_wmma_f32_16x16x4_f32(false, (a), false, (b), (short)0, (c), false, false)

#define HBUF (4 * 16 * 1024 * 64)      // per-tensor head-split buffer, floats
#define OUT_ELEMS (4 * 1024 * 1024)    // out region, floats

// ---------------------------------------------------------------------------
// Kernel 1: Y = X @ W^T + bias, stored head-split as [B, H, S, 64]
// Block tile 128(M) x 64(N), K-chunks of 32. 8 waves: wave w -> rows w*16..+15.
// ---------------------------------------------------------------------------
__global__ __launch_bounds__(256) void qkv_proj_kernel(
    const float* __restrict__ X, const float* __restrict__ W,
    const float* __restrict__ bias, float* __restrict__ Yh) {
  __shared__ float As[128 * 34];
  __shared__ float Bs[64 * 34];
  const int tid = threadIdx.x;
  const int lane = tid & 31;
  const int w = tid >> 5;
  const int n0 = blockIdx.x * 64;
  const int m0 = blockIdx.y * 128;

  v8f acc[4] = {};

  for (int k0 = 0; k0 < 1024; k0 += 32) {
    // A tile 128x32
#pragma unroll
    for (int it = 0; it < 4; ++it) {
      int slot = tid + it * 256;
      int row = slot >> 3, c4 = (slot & 7) << 2;
      float4 f = *(const float4*)(X + (m0 + row) * 1024 + k0 + c4);
      float* d = &As[row * 34 + c4];
      d[0] = f.x; d[1] = f.y; d[2] = f.z; d[3] = f.w;
    }
    // B tile: Bs[n][k] = W[n0+n][k0+k]  (64x32)
#pragma unroll
    for (int it = 0; it < 2; ++it) {
      int slot = tid + it * 256;
      int row = slot >> 3, c4 = (slot & 7) << 2;
      float4 f = *(const float4*)(W + (n0 + row) * 1024 + k0 + c4);
      float* d = &Bs[row * 34 + c4];
      d[0] = f.x; d[1] = f.y; d[2] = f.z; d[3] = f.w;
    }
    __syncthreads();
#pragma unroll
    for (int kk = 0; kk < 32; kk += 4) {
      int kc = kk + ((lane >> 4) << 1);
      int ar = w * 16 + (lane & 15);
      v2f a;
      a.x = As[ar * 34 + kc];
      a.y = As[ar * 34 + kc + 1];
#pragma unroll
      for (int ns = 0; ns < 4; ++ns) {
        int br = ns * 16 + (lane & 15);
        v2f b;
        b.x = Bs[br * 34 + kc];
        b.y = Bs[br * 34 + kc + 1];
        acc[ns] = WMMA_F32(a, b, acc[ns]);
      }
    }
    __syncthreads();
  }
  // epilogue: bias + head-split store
#pragma unroll
  for (int ns = 0; ns < 4; ++ns) {
    int n = n0 + ns * 16 + (lane & 15);
    float bv = bias[n];
    int h = n >> 6, dd = n & 63;
#pragma unroll
    for (int r = 0; r < 8; ++r) {
      int m = m0 + w * 16 + r + ((lane >> 4) << 3);
      int bidx = m >> 10, s = m & 1023;
      Yh[(size_t)(((bidx << 4) + h) * 1024 + s) * 64 + dd] = acc[ns][r] + bv;
    }
  }
}

// ---------------------------------------------------------------------------
// Kernel 2a: raw scores = qh @ kh^T * 0.125 for lower-trapezoid blocks only.
// Block tile 64(i) x 128(j); 8 waves: wave = (row strip rs = w&3) x (col half
// ch = w>>2). K = 64 (dk), loaded once.
// ---------------------------------------------------------------------------
__global__ __launch_bounds__(256) void scores_kernel(
    const float* __restrict__ qh, const float* __restrict__ kh,
    float* __restrict__ scores) {
  __shared__ float Qs[64 * 66];
  __shared__ float Ks[128 * 66];
  const int tid = threadIdx.x;
  const int lane = tid & 31;
  const int w = tid >> 5;
  const int rs = w & 3;
  const int ch = w >> 2;
  const int bh = blockIdx.y;

  // linear block index -> (rb, cb) with cb <= rb/2 (64-row x 128-col blocks)
  int t = blockIdx.x, rb = 0;
  while (t >= (rb >> 1) + 1) { t -= (rb >> 1) + 1; rb++; }
  const int cb = t;
  const int i0 = rb * 64, j0 = cb * 128;

  const float* qb = qh + (size_t)bh * (1024 * 64);
  const float* kb = kh + (size_t)bh * (1024 * 64);

#pragma unroll
  for (int it = 0; it < 4; ++it) {  // Q 64x64
    int slot = tid + it * 256;
    int row = slot >> 4, c4 = (slot & 15) << 2;
    float4 f = *(const float4*)(qb + (i0 + row) * 64 + c4);
    float* d = &Qs[row * 66 + c4];
    d[0] = f.x; d[1] = f.y; d[2] = f.z; d[3] = f.w;
  }
#pragma unroll
  for (int it = 0; it < 8; ++it) {  // K 128x64
    int slot = tid + it * 256;
    int row = slot >> 4, c4 = (slot & 15) << 2;
    float4 f = *(const float4*)(kb + (j0 + row) * 64 + c4);
    float* d = &Ks[row * 66 + c4];
    d[0] = f.x; d[1] = f.y; d[2] = f.z; d[3] = f.w;
  }
  __syncthreads();

  v8f acc[4] = {};
#pragma unroll
  for (int kk = 0; kk < 64; kk += 4) {
    int kc = kk + ((lane >> 4) << 1);
    int ar = rs * 16 + (lane & 15);
    v2f a;
    a.x = Qs[ar * 66 + kc];
    a.y = Qs[ar * 66 + kc + 1];
#pragma unroll
    for (int ns = 0; ns < 4; ++ns) {
      int br = ch * 64 + ns * 16 + (lane & 15);
      v2f b;
      b.x = Ks[br * 66 + kc];
      b.y = Ks[br * 66 + kc + 1];
      acc[ns] = WMMA_F32(a, b, acc[ns]);
    }
  }

  float* sb = scores + (size_t)bh * (1024 * 1024);
#pragma unroll
  for (int ns = 0; ns < 4; ++ns) {
    int j = j0 + ch * 64 + ns * 16 + (lane & 15);
#pragma unroll
    for (int r = 0; r < 8; ++r) {
      int i = i0 + rs * 16 + r + ((lane >> 4) << 3);
      sb[(size_t)i * 1024 + j] = acc[ns][r] * 0.125f;
    }
  }
}

// ---------------------------------------------------------------------------
// Kernel 2b: per-row causal softmax in place; writes exact 0 for j > i.
// One wave32 per row; 8 rows per 256-thread block.
// ---------------------------------------------------------------------------
__global__ __launch_bounds__(256) void softmax_kernel(float* __restrict__ scores) {
  const int lane = threadIdx.x & 31;
  const int w = threadIdx.x >> 5;
  const int rowId = blockIdx.x * 8 + w;
  const int bh = rowId >> 10;
  const int i = rowId & 1023;
  float* p = scores + (size_t)bh * (1024 * 1024) + (size_t)i * 1024;
  const int L = i + 1;

  float e[32];
  float m = -3.402823466e+38f;
#pragma unroll
  for (int it = 0; it < 32; ++it) {
    int j = lane + (it << 5);
    float v = (j < L) ? p[j] : -3.402823466e+38f;
    e[it] = v;
    m = fmaxf(m, v);
  }
#pragma unroll
  for (int off = 16; off > 0; off >>= 1) m = fmaxf(m, __shfl_xor(m, off, 32));
  float s = 0.f;
#pragma unroll
  for (int it = 0; it < 32; ++it) {
    int j = lane + (it << 5);
    float v = (j < L) ? __expf(e[it] - m) : 0.f;
    e[it] = v;
    s += v;
  }
#pragma unroll
  for (int off = 16; off > 0; off >>= 1) s += __shfl_xor(s, off, 32);
  float inv = 1.0f / s;
#pragma unroll
  for (int it = 0; it < 32; ++it) {
    int j = lane + (it << 5);
    p[j] = e[it] * inv;  // e[it]==0 for j>=L -> masked entries become 0
  }
}

// ---------------------------------------------------------------------------
// Kernel 3: out = probs @ vh, head-merge store. Block tile 128(i) x 64(d);
// K loop over j only up to causal limit of the row block.
// ---------------------------------------------------------------------------
__global__ __launch_bounds__(256) void out_kernel(
    const float* __restrict__ probs, const float* __restrict__ vh,
    float* __restrict__ out) {
  __shared__ float Ps[128 * 66];
  __shared__ float Vs[64 * 66];  // transposed: Vs[d][j]
  const int tid = threadIdx.x;
  const int lane = tid & 31;
  const int w = tid >> 5;
  const int bh = blockIdx.y;
  const int rb = blockIdx.x;
  const int i0 = rb * 128;
  const float* pb = probs + (size_t)bh * (1024 * 1024);
  const float* vb = vh + (size_t)bh * (1024 * 64);

  v8f acc[4] = {};
  const int kchunks = (rb + 1) * 2;  // j <= rb*128+127
  for (int kc0 = 0; kc0 < kchunks; ++kc0) {
    int k0 = kc0 * 64;
#pragma unroll
    for (int it = 0; it < 8; ++it) {  // P 128x64
      int slot = tid + it * 256;
      int row = slot >> 4, c4 = (slot & 15) << 2;
      float4 f = *(const float4*)(pb + (size_t)(i0 + row) * 1024 + k0 + c4);
      float* d = &Ps[row * 66 + c4];
      d[0] = f.x; d[1] = f.y; d[2] = f.z; d[3] = f.w;
    }
#pragma unroll
    for (int it = 0; it < 4; ++it) {  // V 64x64 transposed into Vs[d][j]
      int slot = tid + it * 256;
      int j = slot >> 4, d4 = (slot & 15) << 2;
      float4 f = *(const float4*)(vb + (k0 + j) * 64 + d4);
      Vs[(d4 + 0) * 66 + j] = f.x;
      Vs[(d4 + 1) * 66 + j] = f.y;
      Vs[(d4 + 2) * 66 + j] = f.z;
      Vs[(d4 + 3) * 66 + j] = f.w;
    }
    __syncthreads();
#pragma unroll
    for (int kk = 0; kk < 64; kk += 4) {
      int kc = kk + ((lane >> 4) << 1);
      int ar = w * 16 + (lane & 15);
      v2f a;
      a.x = Ps[ar * 66 + kc];
      a.y = Ps[ar * 66 + kc + 1];
#pragma unroll
      for (int ns = 0; ns < 4; ++ns) {
        int br = ns * 16 + (lane & 15);
        v2f b;
        b.x = Vs[br * 66 + kc];
        b.y = Vs[br * 66 + kc + 1];
        acc[ns] = WMMA_F32(a, b, acc[ns]);
      }
    }
    __syncthreads();
  }
  const int b = bh >> 4, h = bh & 15;
#pragma unroll
  for (int ns = 0; ns < 4; ++ns) {
    int dd = ns * 16 + (lane & 15);
#pragma unroll
    for (int r = 0; r < 8; ++r) {
      int i = i0 + w * 16 + r + ((lane >> 4) << 3);
      out[(size_t)((b << 10) + i) * 1024 + (h << 6) + dd] = acc[ns][r];
    }
  }
}

// ---------------------------------------------------------------------------
extern "C" void kernel_launch(void* const* d_in, const int* in_sizes, int n_in,
                              void* d_out, int out_size, void* d_ws, size_t ws_size,
                              hipStream_t stream) {
  (void)in_sizes; (void)n_in; (void)out_size; (void)ws_size;
  const float* q  = (const float*)d_in[0];
  const float* k  = (const float*)d_in[1];
  const float* v  = (const float*)d_in[2];
  // d_in[3] = causal mask (int), handled analytically
  const float* Wq = (const float*)d_in[4];
  const float* bq = (const float*)d_in[5];
  const float* Wk = (const float*)d_in[6];
  const float* bk = (const float*)d_in[7];
  const float* Wv = (const float*)d_in[8];
  const float* bv = (const float*)d_in[9];

  float* out    = (float*)d_out;
  float* scores = out + OUT_ELEMS;          // [B,H,S,S] f32
  float* qh = (float*)d_ws;                 // [B,H,S,64]
  float* kh = qh + HBUF;
  float* vh = kh + HBUF;

  dim3 gProj(16, 32);  // N/64, M/128
  qkv_proj_kernel<<<gProj, 256, 0, stream>>>(q, Wq, bq, qh);
  qkv_proj_kernel<<<gProj, 256, 0, stream>>>(k, Wk, bk, kh);
  qkv_proj_kernel<<<gProj, 256, 0, stream>>>(v, Wv, bv, vh);

  scores_kernel<<<dim3(72, 64), 256, 0, stream>>>(qh, kh, scores);
  softmax_kernel<<<8192, 256, 0, stream>>>(scores);
  out_kernel<<<dim3(8, 64), 256, 0, stream>>>(scores, vh, out);
}